// MultiHeadAttention_1537598292288
// MI455X (gfx1250) — compile-verified
//
#include <hip/hip_runtime.h>
#include <hip/hip_bf16.h>

// MHA forward for MI455X (gfx1250), f32 end-to-end using V_WMMA_F32_16X16X4_F32.
// B=4, S=2048, D=1024, H=16, DK=64.
// v3: scores+softmax fused (attn tensor written exactly once -> halves the
// dominant HBM traffic); 32x64 double-buffered GEMM tiles elsewhere.

typedef float v2f __attribute__((ext_vector_type(2)));
typedef float v8f __attribute__((ext_vector_type(8)));

#define B_  4
#define S_  2048
#define D_  1024
#define H_  16
#define DK_ 64

#define WMMA_F32(A, Bv, C) \
    __builtin_amdgcn_wmma_f32_16x16x4_f32(false, (A), false, (Bv), (short)0, (C), false, false)

// ---------------------------------------------------------------------------
// Y[8192,1024] = X[8192,1024] @ W[1024,1024]^T + bias  (row-major, K contig)
// One wave -> 32x64 tile: 2 A-frags x 4 B-frags -> 8 WMMAs per K-step of 4,
// software double-buffered.
// ---------------------------------------------------------------------------
__global__ void __launch_bounds__(256)
gemm_xwt_bias(const float* __restrict__ X, const float* __restrict__ W,
              const float* __restrict__ bias, float* __restrict__ Y)
{
    const int lane = threadIdx.x & 31;
    const int wave = threadIdx.x >> 5;
    const int wid  = blockIdx.x * 8 + wave;
    const int tn   = wid & 15;               // 1024/64 = 16 col tiles
    const int tm   = wid >> 4;               // 8192/32 = 256 row tiles
    const int m0   = tm * 32;
    const int n0   = tn * 64;
    const int lr   = lane & 15;
    const int lh   = lane >> 4;

    const float* ap0 = X + (size_t)(m0 + lr) * D_ + 2 * lh;
    const float* ap1 = ap0 + (size_t)16 * D_;
    const float* bp  = W + (size_t)(n0 + lr) * D_ + 2 * lh;
    const size_t bstep = (size_t)16 * D_;

    v8f acc[2][4] = {};

    v2f a0c = *(const v2f*)ap0;
    v2f a1c = *(const v2f*)ap1;
    v2f bc[4];
#pragma unroll
    for (int t = 0; t < 4; ++t) bc[t] = *(const v2f*)(bp + (size_t)t * bstep);

    for (int k = 0; k < D_ - 4; k += 4) {
        v2f a0n = *(const v2f*)(ap0 + k + 4);
        v2f a1n = *(const v2f*)(ap1 + k + 4);
        v2f bn[4];
#pragma unroll
        for (int t = 0; t < 4; ++t)
            bn[t] = *(const v2f*)(bp + (size_t)t * bstep + k + 4);

#pragma unroll
        for (int t = 0; t < 4; ++t) {
            acc[0][t] = WMMA_F32(a0c, bc[t], acc[0][t]);
            acc[1][t] = WMMA_F32(a1c, bc[t], acc[1][t]);
        }
        a0c = a0n; a1c = a1n;
#pragma unroll
        for (int t = 0; t < 4; ++t) bc[t] = bn[t];
    }
#pragma unroll
    for (int t = 0; t < 4; ++t) {
        acc[0][t] = WMMA_F32(a0c, bc[t], acc[0][t]);
        acc[1][t] = WMMA_F32(a1c, bc[t], acc[1][t]);
    }

#pragma unroll
    for (int t = 0; t < 4; ++t) {
        const int n = n0 + 16 * t + lr;
        const float bv = bias[n];
#pragma unroll
        for (int g = 0; g < 2; ++g)
#pragma unroll
            for (int r = 0; r < 8; ++r) {
                const int m = m0 + 16 * g + r + 8 * lh;   // C/D: lane half -> M+8
                Y[(size_t)m * D_ + n] = acc[g][t][r] + bv;
            }
    }
}

// ---------------------------------------------------------------------------
// Fused scores+softmax: one 512-thread block (16 waves) owns a full 16x2048
// score strip of one (b,h). Wave w computes cols [w*128, w*128+128) via 8
// WMMA n-tiles (K=64, double-buffered). Row max / exp-sum are reduced with
// __shfl_xor inside each 16-lane half (C-layout: row m = vgpr + 8*laneHalf),
// then combined across waves through a 16x16 LDS array. Normalized weights
// are written to global exactly once.
// ---------------------------------------------------------------------------
__global__ void __launch_bounds__(512)
attn_scores_softmax(const float* __restrict__ Yq, const float* __restrict__ Yk,
                    float* __restrict__ attn)
{
    __shared__ float part[16][16];           // [row][wave] partials
    __shared__ float rowval[16];             // per-row max, then per-row sum

    const int lane = threadIdx.x & 31;
    const int wv   = threadIdx.x >> 5;       // 0..15, also the column strip
    int bid = blockIdx.x;
    const int tm = bid & 127;  bid >>= 7;    // 2048/16 = 128 row tiles
    const int bh = bid;                      // 0..63
    const int bb = bh >> 4, h = bh & 15;
    const int lr = lane & 15, lh = lane >> 4;
    const int n0 = wv * 128;

    const float* qp = Yq + (size_t)(bb * S_ + tm * 16 + lr) * D_ + h * DK_ + 2 * lh;
    const float* kp = Yk + (size_t)(bb * S_ + n0 + lr) * D_ + h * DK_ + 2 * lh;
    const size_t bstep = (size_t)16 * D_;

    v8f acc[8] = {};

    v2f ac = *(const v2f*)qp;
    v2f bc[8];
#pragma unroll
    for (int t = 0; t < 8; ++t) bc[t] = *(const v2f*)(kp + (size_t)t * bstep);

#pragma unroll 5
    for (int k = 0; k < DK_ - 4; k += 4) {
        v2f an = *(const v2f*)(qp + k + 4);
        v2f bn[8];
#pragma unroll
        for (int t = 0; t < 8; ++t)
            bn[t] = *(const v2f*)(kp + (size_t)t * bstep + k + 4);
#pragma unroll
        for (int t = 0; t < 8; ++t) acc[t] = WMMA_F32(ac, bc[t], acc[t]);
        ac = an;
#pragma unroll
        for (int t = 0; t < 8; ++t) bc[t] = bn[t];
    }
#pragma unroll
    for (int t = 0; t < 8; ++t) acc[t] = WMMA_F32(ac, bc[t], acc[t]);

    // ---- scale, then softmax over each 2048-wide row --------------------
    const float scale = 0.125f;              // 1/sqrt(64)
#pragma unroll
    for (int t = 0; t < 8; ++t)
#pragma unroll
        for (int r = 0; r < 8; ++r) acc[t][r] *= scale;

    // per-lane, per-row max over this wave's 8 tiles
    float rmax[8];
#pragma unroll
    for (int r = 0; r < 8; ++r) {
        float m = acc[0][r];
#pragma unroll
        for (int t = 1; t < 8; ++t) m = fmaxf(m, acc[t][r]);
        rmax[r] = m;
    }
    // reduce across the 16 lanes of this half (xor<16 stays in the half)
#pragma unroll
    for (int o = 1; o < 16; o <<= 1)
#pragma unroll
        for (int r = 0; r < 8; ++r) rmax[r] = fmaxf(rmax[r], __shfl_xor(rmax[r], o, 32));
    if (lr == 0) {
#pragma unroll
        for (int r = 0; r < 8; ++r) part[r + 8 * lh][wv] = rmax[r];
    }
    __syncthreads();
    if (wv == 0 && lane < 16) {
        float m = part[lane][0];
#pragma unroll
        for (int i = 1; i < 16; ++i) m = fmaxf(m, part[lane][i]);
        rowval[lane] = m;
    }
    __syncthreads();
    float rm[8];
#pragma unroll
    for (int r = 0; r < 8; ++r) rm[r] = rowval[r + 8 * lh];

    // exp + per-row partial sums
    float rsum[8] = {0.f, 0.f, 0.f, 0.f, 0.f, 0.f, 0.f, 0.f};
#pragma unroll
    for (int t = 0; t < 8; ++t)
#pragma unroll
        for (int r = 0; r < 8; ++r) {
            const float e = __expf(acc[t][r] - rm[r]);
            acc[t][r] = e;
            rsum[r] += e;
        }
#pragma unroll
    for (int o = 1; o < 16; o <<= 1)
#pragma unroll
        for (int r = 0; r < 8; ++r) rsum[r] += __shfl_xor(rsum[r], o, 32);
    if (lr == 0) {
#pragma unroll
        for (int r = 0; r < 8; ++r) part[r + 8 * lh][wv] = rsum[r];
    }
    __syncthreads();
    if (wv == 0 && lane < 16) {
        float s = part[lane][0];
#pragma unroll
        for (int i = 1; i < 16; ++i) s += part[lane][i];
        rowval[lane] = s;
    }
    __syncthreads();
    float inv[8];
#pragma unroll
    for (int r = 0; r < 8; ++r) inv[r] = 1.0f / rowval[r + 8 * lh];

    // ---- single write of normalized attention weights -------------------
    const size_t base = (size_t)bh * S_ * S_;
#pragma unroll
    for (int t = 0; t < 8; ++t) {
        const int n = n0 + 16 * t + lr;
#pragma unroll
        for (int r = 0; r < 8; ++r) {
            const int qi = tm * 16 + r + 8 * lh;
            attn[base + (size_t)qi * S_ + n] = acc[t][r] * inv[r];
        }
    }
}

// ---------------------------------------------------------------------------
// ctx[b,q,h*64+n] = sum_k attn[b,h,q,k] * Vh[b,k,h*64+n]
// Per (b,h): M=2048, N=64, K=2048. One wave -> 32x64 tile (full head width);
// the strided V-fragment loads are shared across both M-tiles; double-buffered.
// ---------------------------------------------------------------------------
__global__ void __launch_bounds__(256)
attn_av(const float* __restrict__ attn, const float* __restrict__ Yv,
        float* __restrict__ ctx)
{
    const int lane = threadIdx.x & 31;
    const int wave = threadIdx.x >> 5;
    int wid = blockIdx.x * 8 + wave;
    const int tm = wid & 63;  wid >>= 6;     // 2048/32 = 64 row tiles
    const int bh = wid;                      // 0..63
    const int bb = bh >> 4, h = bh & 15;
    const int lr = lane & 15, lh = lane >> 4;

    const float* ap0 = attn + (size_t)bh * S_ * S_ + (size_t)(tm * 32 + lr) * S_ + 2 * lh;
    const float* ap1 = ap0 + (size_t)16 * S_;
    const float* vb  = Yv + (size_t)bb * S_ * D_ + h * DK_ + 2 * lh * D_;

    v8f acc[2][4] = {};

    v2f a0c = *(const v2f*)ap0;
    v2f a1c = *(const v2f*)ap1;
    v2f bc[4];
#pragma unroll
    for (int t = 0; t < 4; ++t) {
        const int n = 16 * t + lr;
        bc[t].x = vb[n];
        bc[t].y = vb[D_ + n];
    }

    for (int k0 = 0; k0 < S_ - 4; k0 += 4) {
        v2f a0n = *(const v2f*)(ap0 + k0 + 4);
        v2f a1n = *(const v2f*)(ap1 + k0 + 4);
        const float* vrn = vb + (size_t)(k0 + 4) * D_;
        v2f bn[4];
#pragma unroll
        for (int t = 0; t < 4; ++t) {
            const int n = 16 * t + lr;
            bn[t].x = vrn[n];
            bn[t].y = vrn[D_ + n];
        }

#pragma unroll
        for (int t = 0; t < 4; ++t) {
            acc[0][t] = WMMA_F32(a0c, bc[t], acc[0][t]);
            acc[1][t] = WMMA_F32(a1c, bc[t], acc[1][t]);
        }
        a0c = a0n; a1c = a1n;
#pragma unroll
        for (int t = 0; t < 4; ++t) bc[t] = bn[t];
    }
#pragma unroll
    for (int t = 0; t < 4; ++t) {
        acc[0][t] = WMMA_F32(a0c, bc[t], acc[0][t]);
        acc[1][t] = WMMA_F32(a1c, bc[t], acc[1][t]);
    }

#pragma unroll
    for (int t = 0; t < 4; ++t) {
        const int n = 16 * t + lr;
#pragma unroll
        for (int g = 0; g < 2; ++g)
#pragma unroll
            for (int r = 0; r < 8; ++r) {
                const int qi = tm * 32 + 16 * g + r + 8 * lh;
                ctx[(size_t)(bb * S_ + qi) * D_ + h * DK_ + n] = acc[g][t][r];
            }
    }
}

// ---------------------------------------------------------------------------
extern "C" void kernel_launch(void* const* d_in, const int* in_sizes, int n_in,
                              void* d_out, int out_size, void* d_ws, size_t ws_size,
                              hipStream_t stream)
{
    const float* q   = (const float*)d_in[0];
    const float* k   = (const float*)d_in[1];
    const float* v   = (const float*)d_in[2];
    const float* w_q = (const float*)d_in[3];
    const float* b_q = (const float*)d_in[4];
    const float* w_k = (const float*)d_in[5];
    const float* b_k = (const float*)d_in[6];
    const float* w_v = (const float*)d_in[7];
    const float* b_v = (const float*)d_in[8];
    const float* w_o = (const float*)d_in[9];
    const float* b_o = (const float*)d_in[10];

    float* out  = (float*)d_out;
    float* attn = out + (size_t)B_ * S_ * D_;          // 8,388,608 floats in

    const size_t MD = (size_t)B_ * S_ * D_;            // 8M floats per buffer
    float* Yq  = (float*)d_ws;
    float* Yk  = Yq + MD;
    float* Yv  = Yk + MD;
    float* ctx = Yq;                                   // Yq dead after scores

    const dim3 blk(256);

    // QKV projections: (8192/32)*(1024/64) = 4096 waves -> 512 blocks
    gemm_xwt_bias<<<512, blk, 0, stream>>>(q, w_q, b_q, Yq);
    gemm_xwt_bias<<<512, blk, 0, stream>>>(k, w_k, b_k, Yk);
    gemm_xwt_bias<<<512, blk, 0, stream>>>(v, w_v, b_v, Yv);

    // fused scores+softmax: 64 bh * 128 row strips = 8192 blocks of 512 thr
    attn_scores_softmax<<<8192, dim3(512), 0, stream>>>(Yq, Yk, attn);

    // AV: 64 bh * 64 tiles = 4096 waves -> 512 blocks
    attn_av<<<512, blk, 0, stream>>>(attn, Yv, ctx);

    // output projection -> d_out
    gemm_xwt_bias<<<512, blk, 0, stream>>>(ctx, w_o, b_o, out);
}